// LTFGW_37254546326110
// MI455X (gfx1250) — compile-verified
//
#include <hip/hip_runtime.h>
#include <math.h>

// ---- problem constants (match reference) ----
#define NN   5000
#define DEGC 16
#define SS   17      // neighborhood size
#define TT   10      // templates
#define MM   10      // template nodes
#define DD   128     // features
#define EPSV 0.1f
#define NOUT 3
#define NSNK 5

typedef __attribute__((ext_vector_type(2))) float v2f;
typedef __attribute__((ext_vector_type(8))) float v8f;

// workspace layout (floats)
#define WS_ALPHA 0      // [1]
#define WS_Q     1      // [100]  q[t][m]
#define WS_LQ    101    // [100]  log q
#define WS_C2    201    // [1000] C2[t][i][j] (softmax over i)
#define WS_HC2   1201   // [100]  hC2[t][i]
#define WS_F2SQ  1301   // [100]  ||F2[t,m]||^2

// -------------------------------------------------------------------
// Kernel 1: node-independent precompute (single block, 128 threads)
// -------------------------------------------------------------------
__global__ __launch_bounds__(128)
void ltfgw_pre(const float* __restrict__ tmpl,   // [T,M,M]
               const float* __restrict__ F2,     // [T,M,D]
               const float* __restrict__ q0,     // [T,M]
               const float* __restrict__ alpha0, // [1]
               float* __restrict__ ws) {
  int tid = threadIdx.x;
  if (tid == 0) ws[WS_ALPHA] = 1.0f / (1.0f + expf(-alpha0[0]));

  // q = softmax(q0, axis=1); log_q
  if (tid < TT) {
    float mx = -__builtin_inff();
    for (int m = 0; m < MM; ++m) mx = fmaxf(mx, q0[tid*MM + m]);
    float se = 0.f;
    for (int m = 0; m < MM; ++m) se += expf(q0[tid*MM + m] - mx);
    float lse = mx + logf(se);
    for (int m = 0; m < MM; ++m) {
      float lq = q0[tid*MM + m] - lse;
      ws[WS_Q  + tid*MM + m] = expf(lq);
      ws[WS_LQ + tid*MM + m] = lq;
    }
  }
  // C2 = softmax(templates, axis=1): per (t, column j), over rows i
  if (tid < TT*MM) {
    int t = tid / MM, j = tid % MM;
    float mx = -__builtin_inff();
    for (int i = 0; i < MM; ++i) mx = fmaxf(mx, tmpl[t*MM*MM + i*MM + j]);
    float se = 0.f;
    for (int i = 0; i < MM; ++i) se += expf(tmpl[t*MM*MM + i*MM + j] - mx);
    float inv = 1.0f / se;
    for (int i = 0; i < MM; ++i)
      ws[WS_C2 + t*MM*MM + i*MM + j] = expf(tmpl[t*MM*MM + i*MM + j] - mx) * inv;
  }
  // ||F2[c]||^2
  if (tid < TT*MM) {
    float s = 0.f;
    for (int d = 0; d < DD; ++d) { float v = F2[tid*DD + d]; s += v*v; }
    ws[WS_F2SQ + tid] = s;
  }
  __syncthreads();
  // hC2[t][i] = sum_j C2[t][i][j]^2 * q[t][j]
  if (tid < TT*MM) {
    int t = tid / MM, i = tid % MM;
    float s = 0.f;
    for (int j = 0; j < MM; ++j) {
      float c = ws[WS_C2 + t*MM*MM + i*MM + j];
      s += c*c*ws[WS_Q + t*MM + j];
    }
    ws[WS_HC2 + tid] = s;
  }
}

// -------------------------------------------------------------------
// Kernel 2: one block per node; 10 waves = 10 templates
// -------------------------------------------------------------------
__global__ __launch_bounds__(320)
void ltfgw_main(const float* __restrict__ x,    // [N,D]
                const int*   __restrict__ dstp, // [N,DEG]
                const float* __restrict__ F2,   // [T*M, D]
                const float* __restrict__ ws,
                float* __restrict__ out) {      // [N,T]
  __shared__ int   nl[SS];
  __shared__ float xl[SS][DD];
  __shared__ int   adjRaw[SS][SS];
  __shared__ float C1[SS][SS];
  __shared__ float hC1[SS];
  __shared__ float xlsq[SS];
  __shared__ float Msm[TT][SS][MM];
  __shared__ float Hsm[TT][SS][MM];
  __shared__ float C2l[TT][MM*MM];
  __shared__ float q_l[TT*MM], lq_l[TT*MM], hC2_l[TT*MM];

  const int n    = blockIdx.x;
  const int tid  = threadIdx.x;
  const int w    = tid >> 5;      // wave id (uniform per wave)
  const int lane = tid & 31;
  const int half = lane >> 4;
  const int l16  = lane & 15;

  // ---- P0: node list + small tables ----
  if (tid < SS) nl[tid] = (tid == 0) ? n : dstp[n*DEGC + (tid - 1)];
  for (int i = tid; i < TT*MM*MM; i += 320) C2l[i/(MM*MM)][i%(MM*MM)] = ws[WS_C2 + i];
  for (int i = tid; i < TT*MM; i += 320) {
    q_l[i]   = ws[WS_Q   + i];
    lq_l[i]  = ws[WS_LQ  + i];
    hC2_l[i] = ws[WS_HC2 + i];
  }
  __syncthreads();

  // ---- P1: gather xl; raw adjacency ----
  for (int i = tid; i < SS*DD; i += 320) {
    int r = i >> 7, c = i & (DD - 1);
    xl[r][c] = x[nl[r]*DD + c];
  }
  for (int i = tid; i < SS*SS; i += 320) {
    int a = i / SS, b = i % SS;
    int na = nl[a], vb = nl[b];
    int hit = 0;
    for (int k = 0; k < DEGC; ++k) hit |= (dstp[na*DEGC + k] == vb) ? 1 : 0;
    adjRaw[a][b] = hit;
  }
  __syncthreads();

  // ---- P2: C1 (symmetrized, no diag), xlsq ----
  for (int i = tid; i < SS*SS; i += 320) {
    int a = i / SS, b = i % SS;
    C1[a][b] = (a != b && (adjRaw[a][b] | adjRaw[b][a])) ? 1.0f : 0.0f;
  }
  if (tid < SS) {
    float s = 0.f;
    for (int d = 0; d < DD; ++d) s += xl[tid][d]*xl[tid][d];
    xlsq[tid] = s;
  }
  __syncthreads();

  // ---- P3: hC1, and M = ||xl-F2||^2 via f32 WMMA ----
  if (tid < SS) {
    float s = 0.f;
    for (int j = 0; j < SS; ++j) s += C1[tid][j];  // C1 is 0/1 => C1^2 == C1
    hC1[tid] = s * (1.0f / SS);
  }
  if (w < 7) {
    // wave w owns output columns [16w, 16w+16) of the 17x100 GEMM (rows 0..15)
    const int colbase = w * 16;
    const int c  = colbase + l16;
    const int cc = (c < TT*MM) ? c : (TT*MM - 1);   // clamp to stay in-bounds
    const float cm = (c < TT*MM) ? 1.0f : 0.0f;
    v8f acc = {};
    for (int k0 = 0; k0 < DD; k0 += 4) {
      v2f a, b;
      // A 16x4 f32 layout: lanes 0-15 -> K=k0,k0+1 ; lanes 16-31 -> K=k0+2,k0+3
      a.x = xl[l16][k0 + 2*half];
      a.y = xl[l16][k0 + 2*half + 1];
      const float* fp = F2 + cc*DD + k0 + 2*half;
      b.x = fp[0] * cm;
      b.y = fp[1] * cm;
      acc = __builtin_amdgcn_wmma_f32_16x16x4_f32(false, a, false, b,
                                                  (short)0, acc, false, false);
    }
    if (c < TT*MM) {
      int t = c / MM, m = c % MM;
      float f2s = ws[WS_F2SQ + c];
      #pragma unroll
      for (int r = 0; r < 8; ++r) {
        int srow = r + 8*half;                     // C/D: VGPR r, lane-half -> row
        Msm[t][srow][m] = xlsq[srow] + f2s - 2.0f*acc[r];
      }
    }
  } else {
    // row s=16 handled scalar by waves 7..9 (96 threads)
    int j = tid - 224;
    for (int c = j; c < TT*MM; c += 96) {
      float s = 0.f;
      for (int d = 0; d < DD; ++d) s += xl[SS-1][d]*F2[c*DD + d];
      Msm[c/MM][SS-1][c%MM] = xlsq[SS-1] + ws[WS_F2SQ + c] - 2.0f*s;
    }
  }
  __syncthreads();

  // ---- P4: per-template FGW outer/Sinkhorn loop (wave t = w) ----
  const int   t    = w;
  const bool  act  = (lane < SS);
  const int   srow = act ? lane : 0;
  const float alpha = ws[WS_ALPHA];
  const float oma   = 1.0f - alpha;
  const float invE  = 1.0f / EPSV;
  const float logp  = -logf((float)SS);
  const float NEG_INF = -__builtin_inff();

  float Mrow[MM], cC[MM], lqt[MM], G[MM];
  {
    float h1 = hC1[srow];
    #pragma unroll
    for (int m = 0; m < MM; ++m) {
      Mrow[m] = Msm[t][srow][m];
      cC[m]   = h1 + hC2_l[t*MM + m];
      lqt[m]  = lq_l[t*MM + m];
      G[m]    = act ? (1.0f/SS) * q_l[t*MM + m] : 0.0f;
    }
  }

  for (int outer = 0; outer <= NOUT; ++outer) {
    // H[s][l] = sum_k G[s][k] * C2[t][l][k]
    #pragma unroll
    for (int l = 0; l < MM; ++l) {
      float h = 0.f;
      #pragma unroll
      for (int k = 0; k < MM; ++k) h += G[k] * C2l[t][l*MM + k];
      if (act) Hsm[t][srow][l] = h;
    }
    __syncthreads();   // uniform: all waves run identical loop structure
    // tens[s][l] = constC - 2 * sum_j C1[s][j] * H[j][l]
    float tens[MM];
    #pragma unroll
    for (int l = 0; l < MM; ++l) {
      float s2 = 0.f;
      for (int j = 0; j < SS; ++j) s2 += C1[srow][j] * Hsm[t][j][l];
      tens[l] = cC[l] - 2.0f*s2;
    }
    if (outer == NOUT) {
      // dist = sum G * ((1-a)M + a*tens)
      float dsum = 0.f;
      if (act) {
        #pragma unroll
        for (int m = 0; m < MM; ++m)
          dsum += G[m] * (oma*Mrow[m] + alpha*tens[m]);
      }
      for (int off = 16; off > 0; off >>= 1) dsum += __shfl_xor(dsum, off, 32);
      if (lane == 0) out[n*TT + t] = dsum;
      break;
    }
    // logK = -((1-a)M + 2a*tens)/eps ; Sinkhorn (a,b reset each outer iter)
    float logK[MM];
    #pragma unroll
    for (int m = 0; m < MM; ++m)
      logK[m] = -(oma*Mrow[m] + 2.0f*alpha*tens[m]) * invE;

    float av = act ? 0.0f : NEG_INF;
    float bv[MM];
    #pragma unroll
    for (int m = 0; m < MM; ++m) bv[m] = 0.f;

    for (int it = 0; it < NSNK; ++it) {
      // a = log_p - LSE_m(logK + b)   (per-lane)
      float mx = NEG_INF;
      #pragma unroll
      for (int m = 0; m < MM; ++m) mx = fmaxf(mx, logK[m] + bv[m]);
      float se = 0.f;
      #pragma unroll
      for (int m = 0; m < MM; ++m) se += __expf(logK[m] + bv[m] - mx);
      av = act ? (logp - (mx + __logf(se))) : NEG_INF;
      // b[m] = log_q - LSE_s(logK + a)   (cross-lane, wave32 shfl reduce)
      #pragma unroll
      for (int m = 0; m < MM; ++m) {
        float v = logK[m] + av;           // -inf on inactive lanes
        float mmx = v;
        for (int off = 16; off > 0; off >>= 1)
          mmx = fmaxf(mmx, __shfl_xor(mmx, off, 32));
        float es = __expf(v - mmx);
        for (int off = 16; off > 0; off >>= 1)
          es += __shfl_xor(es, off, 32);
        bv[m] = lqt[m] - (mmx + __logf(es));
      }
    }
    #pragma unroll
    for (int m = 0; m < MM; ++m)
      G[m] = act ? __expf(logK[m] + av + bv[m]) : 0.0f;
  }
}

extern "C" void kernel_launch(void* const* d_in, const int* in_sizes, int n_in,
                              void* d_out, int out_size, void* d_ws, size_t ws_size,
                              hipStream_t stream) {
  (void)in_sizes; (void)n_in; (void)out_size; (void)ws_size;
  const float* x      = (const float*)d_in[0];
  const int*   edge   = (const int*)  d_in[1];     // [2, N*DEG]
  const float* tmpl   = (const float*)d_in[2];     // [T,M,M]
  const float* tfeat  = (const float*)d_in[3];     // [T,M,D]
  const float* q0     = (const float*)d_in[4];     // [T,M]
  const float* alpha0 = (const float*)d_in[5];     // [1]
  const int*   dstp   = edge + NN*DEGC;            // row 1 = dst
  float*       ws     = (float*)d_ws;
  float*       out    = (float*)d_out;

  ltfgw_pre <<<1, 128, 0, stream>>>(tmpl, tfeat, q0, alpha0, ws);
  ltfgw_main<<<NN, 320, 0, stream>>>(x, dstp, tfeat, ws, out);
}